// QValueAgent_67095979098191
// MI455X (gfx1250) — compile-verified
//
#include <hip/hip_runtime.h>
#include <cstdint>

// Problem geometry (fixed by the reference).
#define B_SESS   4096
#define T_TRIALS 2048
#define CHUNK    64                      // timesteps per chunk
#define NCHUNK   (T_TRIALS / CHUNK)      // 32 chunks
#define ROW_BYTES (CHUNK * 12 + 16)      // 784 B: 64 steps * 3 floats + 16B pad (bank-conflict relief, keeps 16B align)
#define ROW_FLOATS (ROW_BYTES / 4)       // 196

typedef __attribute__((ext_vector_type(4))) float v4f;  // clang vector: valid for nontemporal builtins

// d_ws layout: float4 tuple per (session, chunk):
//   pass1 writes (P_l, S_l, P_r, S_r); pass2 overwrites (.x,.y) with (vl_start, vr_start).
// Uses B_SESS * NCHUNK * 16 = 2 MiB of workspace.

__device__ __forceinline__ float sigmoidf_(float x) {
  return 1.0f / (1.0f + __expf(-x));
}

// CDNA5 async global->LDS DMA of one 32-session x CHUNK-step input tile.
// Each lane copies its session's contiguous 768-byte stream into its padded LDS row.
// IOFFSET is added to BOTH the LDS and the global address (ISA §15.18 async pseudocode),
// which matches our +16B/step pattern exactly -> zero per-iteration address VALU.
// Tracked by ASYNCcnt; wave-private, so no barrier needed (1 wave per workgroup).
template <int K>
__device__ __forceinline__ void stage_rec(uint32_t lds_off, uint32_t gl_off, uint64_t base) {
  if constexpr (K < (CHUNK * 12) / 16) {
    asm volatile("global_load_async_to_lds_b128 %0, %1, %2 offset:%3"
                 :
                 : "v"(lds_off), "v"(gl_off), "s"(base), "i"(K * 16)
                 : "memory");
    stage_rec<K + 1>(lds_off, gl_off, base);
  }
}

__device__ __forceinline__ void stage_tile_async(uint32_t lds_off, uint32_t gl_off,
                                                 uint64_t base) {
  stage_rec<0>(lds_off, gl_off, base);
  asm volatile("s_wait_asynccnt 0x0" ::: "memory");
}

struct Consts {
  float al0, al1, al2, al3;      // sigmoid(a_*)
  float alk0, alk1, alk2, alk3;  // alpha_i * k_i
  float k0, k1, k2, k3;
};

__device__ __forceinline__ Consts load_consts(const float* asr, const float* asu,
                                              const float* adr, const float* adu,
                                              const float* ksr, const float* ksu,
                                              const float* kdr, const float* kdu) {
  Consts c;
  c.al0 = sigmoidf_(asr[0]); c.al1 = sigmoidf_(asu[0]);
  c.al2 = sigmoidf_(adr[0]); c.al3 = sigmoidf_(adu[0]);
  c.k0 = ksr[0]; c.k1 = ksu[0]; c.k2 = kdr[0]; c.k3 = kdu[0];
  c.alk0 = c.al0 * c.k0; c.alk1 = c.al1 * c.k1;
  c.alk2 = c.al2 * c.k2; c.alk3 = c.al3 * c.k3;
  return c;
}

// ---- Pass 1: per (session, chunk) compute linear-recurrence composite (P,S) per side.
__global__ void __launch_bounds__(32)
qv_pass1(const float* __restrict__ inp,
         const float* asr, const float* asu, const float* adr, const float* adu,
         const float* ksr, const float* ksu, const float* kdr, const float* kdu,
         float4* __restrict__ ws) {
  __shared__ float tile[32 * ROW_FLOATS];
  const int lane = threadIdx.x;
  const int c    = blockIdx.x & (NCHUNK - 1);
  const int sg   = blockIdx.x >> 5;        // NCHUNK==32
  const int s    = sg * 32 + lane;

  const uint32_t lds0 = (uint32_t)(uintptr_t)tile + (uint32_t)lane * ROW_BYTES;
  const uint32_t goff = (uint32_t)s * (T_TRIALS * 12u) + (uint32_t)c * (CHUNK * 12u);
  stage_tile_async(lds0, goff, (uint64_t)(uintptr_t)inp);

  const Consts K = load_consts(asr, asu, adr, adu, ksr, ksu, kdr, kdu);

  const float* row = tile + lane * ROW_FLOATS;
  float Pl = 1.0f, Sl = 0.0f, Pr = 1.0f, Sr = 0.0f;
#pragma unroll 8
  for (int j = 0; j < CHUNK; ++j) {
    const float cl = row[j * 3 + 0], cr = row[j * 3 + 1], o = row[j * 3 + 2];
    const float no = 1.0f - o;
    const float p0 = cl * o, p1 = cl * no, p2 = cr * o, p3 = cr * no;
    const float CAl = K.al0 * p0 + K.al1 * p1 + K.al2 * p2 + K.al3 * p3;
    const float KAl = K.alk0 * p0 + K.alk1 * p1 + K.alk2 * p2 + K.alk3 * p3;
    const float CAr = K.al0 * p2 + K.al1 * p3 + K.al2 * p0 + K.al3 * p1;
    const float KAr = K.alk0 * p2 + K.alk1 * p3 + K.alk2 * p0 + K.alk3 * p1;
    const float ml = 1.0f - CAl, mr = 1.0f - CAr;
    Pl *= ml; Sl = fmaf(Sl, ml, KAl);
    Pr *= mr; Sr = fmaf(Sr, mr, KAr);
  }
  ws[s * NCHUNK + c] = make_float4(Pl, Sl, Pr, Sr);
}

// ---- Pass 2: per-session scan over the 32 chunk composites -> v at each chunk start.
__global__ void __launch_bounds__(256)
qv_scan(float4* __restrict__ ws) {
  const int s = blockIdx.x * blockDim.x + threadIdx.x;
  float vl = 0.0f, vr = 0.0f;
  float4* p = ws + s * NCHUNK;
#pragma unroll
  for (int c = 0; c < NCHUNK; ++c) {
    const float4 t = p[c];
    p[c] = make_float4(vl, vr, 0.0f, 0.0f);   // v BEFORE this chunk
    vl = fmaf(vl, t.x, t.y);
    vr = fmaf(vr, t.z, t.w);
  }
}

// One timestep of the replay for both sides; returns (diff_l, diff_r, del_l, del_r)
// and updates vl/vr in place.
__device__ __forceinline__ v4f qv_step(const Consts& K, float cl, float cr, float o,
                                       float& vl, float& vr) {
  const float no = 1.0f - o;
  const float p0 = cl * o, p1 = cl * no, p2 = cr * o, p3 = cr * no;
  const float Cs  = cl + cr;
  const float Ksl = K.k0 * p0 + K.k1 * p1 + K.k2 * p2 + K.k3 * p3;
  const float Ksr = K.k0 * p2 + K.k1 * p3 + K.k2 * p0 + K.k3 * p1;
  const float CAl = K.al0 * p0 + K.al1 * p1 + K.al2 * p2 + K.al3 * p3;
  const float KAl = K.alk0 * p0 + K.alk1 * p1 + K.alk2 * p2 + K.alk3 * p3;
  const float CAr = K.al0 * p2 + K.al1 * p3 + K.al2 * p0 + K.al3 * p1;
  const float KAr = K.alk0 * p2 + K.alk1 * p3 + K.alk2 * p0 + K.alk3 * p1;
  const float diffl = Ksl - vl * Cs,  diffr = Ksr - vr * Cs;   // diff uses OLD v
  const float dell  = KAl - vl * CAl, delr  = KAr - vr * CAr;  // delta uses OLD v
  vl += dell; vr += delr;
  v4f r = {diffl, diffr, dell, delr};
  return r;
}

// ---- Pass 3: replay each chunk from its known v_start, emit (v, diff, delta).
// Outputs are write-once streams -> non-temporal b128 stores so the input's
// L2 residency (96 MB in 192 MB L2, warmed by pass 1) survives for the re-read.
__global__ void __launch_bounds__(32)
qv_pass3(const float* __restrict__ inp, const float4* __restrict__ ws,
         float2* __restrict__ out,  // [v | diff | delta], each B*T float2 entries
         const float* asr, const float* asu, const float* adr, const float* adu,
         const float* ksr, const float* ksu, const float* kdr, const float* kdu) {
  __shared__ float tile[32 * ROW_FLOATS];
  const int lane = threadIdx.x;
  const int c    = blockIdx.x & (NCHUNK - 1);
  const int sg   = blockIdx.x >> 5;
  const int s    = sg * 32 + lane;

  const uint32_t lds0 = (uint32_t)(uintptr_t)tile + (uint32_t)lane * ROW_BYTES;
  const uint32_t goff = (uint32_t)s * (T_TRIALS * 12u) + (uint32_t)c * (CHUNK * 12u);
  stage_tile_async(lds0, goff, (uint64_t)(uintptr_t)inp);

  const Consts K = load_consts(asr, asu, adr, adu, ksr, ksu, kdr, kdu);

  const float4 vs = ws[s * NCHUNK + c];
  float vl = vs.x, vr = vs.y;

  const size_t e0 = (size_t)s * T_TRIALS + (size_t)c * CHUNK;  // float2 index, 16B aligned
  v4f* __restrict__ ov4 = (v4f*)(out + e0);
  v4f* __restrict__ od4 = (v4f*)(out + (size_t)B_SESS * T_TRIALS + e0);
  v4f* __restrict__ ox4 = (v4f*)(out + 2ull * B_SESS * T_TRIALS + e0);

  const float* row = tile + lane * ROW_FLOATS;
#pragma unroll 4
  for (int j = 0; j < CHUNK; j += 2) {
    const float cl0 = row[j * 3 + 0], cr0 = row[j * 3 + 1], o0 = row[j * 3 + 2];
    const float cl1 = row[j * 3 + 3], cr1 = row[j * 3 + 4], o1 = row[j * 3 + 5];

    const v4f r0 = qv_step(K, cl0, cr0, o0, vl, vr);
    const float vl0 = vl, vr0 = vr;
    const v4f r1 = qv_step(K, cl1, cr1, o1, vl, vr);

    const int q = j >> 1;
    const v4f vv = {vl0, vr0, vl, vr};
    const v4f dd = {r0.x, r0.y, r1.x, r1.y};
    const v4f xx = {r0.z, r0.w, r1.z, r1.w};
    __builtin_nontemporal_store(vv, &ov4[q]);  // v     (t, t+1)
    __builtin_nontemporal_store(dd, &od4[q]);  // diff  (t, t+1)
    __builtin_nontemporal_store(xx, &ox4[q]);  // delta (t, t+1)
  }
}

extern "C" void kernel_launch(void* const* d_in, const int* in_sizes, int n_in,
                              void* d_out, int out_size, void* d_ws, size_t ws_size,
                              hipStream_t stream) {
  (void)in_sizes; (void)n_in; (void)out_size; (void)ws_size;
  const float* inp = (const float*)d_in[0];
  const float* asr = (const float*)d_in[1];
  const float* asu = (const float*)d_in[2];
  const float* adr = (const float*)d_in[3];
  const float* adu = (const float*)d_in[4];
  const float* ksr = (const float*)d_in[5];
  const float* ksu = (const float*)d_in[6];
  const float* kdr = (const float*)d_in[7];
  const float* kdu = (const float*)d_in[8];
  float4* ws = (float4*)d_ws;                       // 2 MiB used
  float2* out = (float2*)d_out;

  const int nblk = (B_SESS / 32) * NCHUNK;          // 4096 single-wave workgroups
  qv_pass1<<<nblk, 32, 0, stream>>>(inp, asr, asu, adr, adu, ksr, ksu, kdr, kdu, ws);
  qv_scan<<<B_SESS / 256, 256, 0, stream>>>(ws);
  qv_pass3<<<nblk, 32, 0, stream>>>(inp, ws, out, asr, asu, adr, adu, ksr, ksu, kdr, kdu);
}